// SplitChargeEquilibrationBlock_74028056314327
// MI455X (gfx1250) — compile-verified
//

#include <hip/hip_runtime.h>
#include <hip/hip_bf16.h>
#include <math.h>
#include <stdint.h>

// ---------------------------------------------------------------------------
// SplitChargeEquilibration on MI455X (gfx1250).
//   A[64x64] (erf kernel)  ->  Ap = T^T A T (WMMA f32 16x16x4)
//   -> augmented 513x513 solve via blocked (NB=16) unpivoted LU with
//      WMMA f32 trailing updates; U12 block row staged into LDS by the
//      Tensor Data Mover (tensor_load_to_lds + s_wait_tensorcnt)
//   -> back-substitution -> charges = T @ x.
// One workgroup (8 wave32) per molecule for the LU: no inter-block sync.
// ---------------------------------------------------------------------------

#define BATCH 256
#define NN    64      // atoms per molecule
#define EE    512     // edges per molecule
#define NUM_EL 10
#define NPAD  528     // padded row count (33 * 16), real system is 513
#define NCOLP 544     // padded col count (34 * 16); col 528 = b vector
#define LDA   544     // row stride (floats) of Aaug
#define R_CUT 10.0f

typedef float v2f __attribute__((ext_vector_type(2)));
typedef float v8f __attribute__((ext_vector_type(8)));
typedef unsigned int v4u __attribute__((ext_vector_type(4)));
typedef int v8i __attribute__((ext_vector_type(8)));
typedef int v4i __attribute__((ext_vector_type(4)));

// --------------------------------------------------------------------------
// Kernel 1: build A[64x64] per molecule.
// --------------------------------------------------------------------------
__global__ void k_build_A(const float* __restrict__ pos,
                          const float* __restrict__ node_attrs,
                          const int*   __restrict__ an,
                          const float* __restrict__ hardness,
                          const float* __restrict__ cov,
                          float* __restrict__ wsA) {
  int mol = blockIdx.x;
  int tid = threadIdx.x;
  __shared__ float sx[NN], sy[NN], sz[NN], sr[NN], sdiag[NN];
  if (tid < NN) {
    const float* P = pos + (size_t)mol * NN * 3;
    sx[tid] = P[tid * 3 + 0];
    sy[tid] = P[tid * 3 + 1];
    sz[tid] = P[tid * 3 + 2];
    float r = cov[an[mol * NN + tid]];
    sr[tid] = r;
    const float* na = node_attrs + ((size_t)mol * NN + tid) * NUM_EL;
    int am = 0; float bv = na[0];
    #pragma unroll
    for (int j = 1; j < NUM_EL; ++j) { float v = na[j]; if (v > bv) { bv = v; am = j; } }
    sdiag[tid] = hardness[am] + 1.0f / (1.7724538509055159f * r);  // 1/(sqrt(pi)*r)
  }
  __syncthreads();
  float* Aout = wsA + (size_t)mol * NN * NN;
  const float inv_sqrt2 = 0.70710678118654752f;
  for (int idx = tid; idx < NN * NN; idx += blockDim.x) {
    int i = idx / NN, j = idx % NN;
    float val;
    if (i == j) {
      val = sdiag[i];
    } else {
      float dx = sx[i] - sx[j], dy = sy[i] - sy[j], dz = sz[i] - sz[j];
      float d = sqrtf(dx * dx + dy * dy + dz * dz);
      float gamma = sqrtf(sr[i] * sr[i] + sr[j] * sr[j]);
      val = erff(d * inv_sqrt2 / gamma) / d;
    }
    Aout[idx] = val;
  }
}

// --------------------------------------------------------------------------
// Kernel 2: per-edge cutoff diagonal and bvec = -(T^T eneg).
// --------------------------------------------------------------------------
__global__ void k_edge_b(const float* __restrict__ pos,
                         const float* __restrict__ T,
                         const float* __restrict__ eneg,
                         const int*   __restrict__ eidx,
                         float* __restrict__ wsCut,
                         float* __restrict__ wsBvec) {
  int mol = blockIdx.x;
  int e = threadIdx.x;                 // blockDim = 512
  const float* P = pos + (size_t)mol * NN * 3;
  int e0 = eidx[(size_t)mol * 2 * EE + e];
  int e1 = eidx[(size_t)mol * 2 * EE + EE + e];
  float dx = P[e0 * 3 + 0] - P[e1 * 3 + 0];
  float dy = P[e0 * 3 + 1] - P[e1 * 3 + 1];
  float dz = P[e0 * 3 + 2] - P[e1 * 3 + 2];
  float d = sqrtf(dx * dx + dy * dy + dz * dz);
  wsCut[(size_t)mol * EE + e] =
      1.0f / cosf(3.14159265358979323846f * d / (2.0f * R_CUT)) - 1.0f;
  const float* Tm = T + (size_t)mol * NN * EE;
  __shared__ float sen[NN];
  if (e < NN) sen[e] = eneg[(size_t)mol * NN + e];
  __syncthreads();
  float acc = 0.0f;
  #pragma unroll 4
  for (int n = 0; n < NN; ++n) acc += Tm[(size_t)n * EE + e] * sen[n];
  wsBvec[(size_t)mol * EE + e] = -acc;
}

// --------------------------------------------------------------------------
// Kernel 3: M = A @ T   [64 x 512], f32 WMMA 16x16x4. A staged in LDS.
// --------------------------------------------------------------------------
__global__ void k_gemm_AT(const float* __restrict__ T,
                          const float* __restrict__ wsA,
                          float* __restrict__ wsM) {
  int mol = blockIdx.x;
  int tid = threadIdx.x;               // blockDim = 256 (8 waves)
  __shared__ float sA[NN * NN];
  for (int i = tid; i < NN * NN; i += blockDim.x)
    sA[i] = wsA[(size_t)mol * NN * NN + i];
  __syncthreads();
  const float* Tm = T + (size_t)mol * NN * EE;
  float* Mm = wsM + (size_t)mol * NN * EE;
  int wave = tid >> 5, l = tid & 31, lm = l & 15, half = l >> 4;
  const int NT_R = NN / 16, NT_C = EE / 16;   // 4 x 32 tiles
  for (int t = wave; t < NT_R * NT_C; t += 8) {
    int i0 = (t / NT_C) * 16, c0 = (t % NT_C) * 16;
    v8f acc = {0.f,0.f,0.f,0.f,0.f,0.f,0.f,0.f};
    #pragma unroll 4
    for (int ks = 0; ks < NN; ks += 4) {
      int k0 = ks + half * 2;
      v2f a, b;
      a.x = sA[(i0 + lm) * NN + k0];
      a.y = sA[(i0 + lm) * NN + k0 + 1];
      b.x = Tm[(size_t)(k0)     * EE + c0 + lm];
      b.y = Tm[(size_t)(k0 + 1) * EE + c0 + lm];
      acc = __builtin_amdgcn_wmma_f32_16x16x4_f32(false, a, false, b,
                                                  (short)0, acc, false, false);
    }
    #pragma unroll
    for (int v = 0; v < 8; ++v)
      Mm[(size_t)(i0 + half * 8 + v) * EE + c0 + lm] = acc[v];
  }
}

// --------------------------------------------------------------------------
// Kernel 4: Aaug core = T^T @ M + diag(cut), WMMA f32, into padded Aaug.
// grid = (BATCH, 4); block handles 8 tile-rows of 32 tiles.
// --------------------------------------------------------------------------
__global__ void k_gemm_TtM(const float* __restrict__ T,
                           const float* __restrict__ wsM,
                           const float* __restrict__ wsCut,
                           float* __restrict__ wsAaug) {
  int mol = blockIdx.x, quarter = blockIdx.y;
  const float* Tm  = T   + (size_t)mol * NN * EE;
  const float* Mm  = wsM + (size_t)mol * NN * EE;
  const float* cut = wsCut + (size_t)mol * EE;
  float* Ag = wsAaug + (size_t)mol * NPAD * LDA;
  int tid = threadIdx.x;               // 256
  int wave = tid >> 5, l = tid & 31, lm = l & 15, half = l >> 4;
  for (int t = wave; t < 8 * 32; t += 8) {
    int tr = 8 * quarter + t / 32, tc = t % 32;
    int e0 = tr * 16, f0 = tc * 16;
    v8f acc = {0.f,0.f,0.f,0.f,0.f,0.f,0.f,0.f};
    #pragma unroll 4
    for (int ks = 0; ks < NN; ks += 4) {
      int k0 = ks + half * 2;
      v2f a, b;
      a.x = Tm[(size_t)(k0)     * EE + e0 + lm];
      a.y = Tm[(size_t)(k0 + 1) * EE + e0 + lm];
      b.x = Mm[(size_t)(k0)     * EE + f0 + lm];
      b.y = Mm[(size_t)(k0 + 1) * EE + f0 + lm];
      acc = __builtin_amdgcn_wmma_f32_16x16x4_f32(false, a, false, b,
                                                  (short)0, acc, false, false);
    }
    #pragma unroll
    for (int v = 0; v < 8; ++v) {
      int r = e0 + half * 8 + v, c = f0 + lm;
      float val = acc[v] + ((r == c) ? cut[r] : 0.0f);
      Ag[(size_t)r * LDA + c] = val;
    }
  }
}

// --------------------------------------------------------------------------
// Kernel 5: augmented borders + padding + b column (col 528).
// --------------------------------------------------------------------------
__global__ void k_finish_aaug(const float* __restrict__ wsBvec,
                              const float* __restrict__ qtot,
                              float* __restrict__ wsAaug) {
  int mol = blockIdx.x;
  int tid = threadIdx.x;               // 256
  float* Ag = wsAaug + (size_t)mol * NPAD * LDA;
  const float* bv = wsBvec + (size_t)mol * EE;
  // rows 0..511: col 512 = 1; cols 513..527 = 0; col 528 = b; 529..543 = 0
  for (int r = tid; r < EE; r += blockDim.x) {
    float* row = Ag + (size_t)r * LDA;
    row[512] = 1.0f;
    #pragma unroll
    for (int c = 513; c < 528; ++c) row[c] = 0.0f;
    row[528] = bv[r];
    #pragma unroll
    for (int c = 529; c < NCOLP; ++c) row[c] = 0.0f;
  }
  // row 512: ones over 0..511, 0 at 512, 0 padding, q_tot at 528
  {
    float* row = Ag + (size_t)512 * LDA;
    for (int c = tid; c < NCOLP; c += blockDim.x) {
      float v = 0.0f;
      if (c < 512) v = 1.0f;
      else if (c == 528) v = qtot[mol];
      row[c] = v;
    }
  }
  // padding rows 513..527: unit diagonal, zeros elsewhere
  for (int idx = tid; idx < (NPAD - 513) * NCOLP; idx += blockDim.x) {
    int r = 513 + idx / NCOLP, c = idx % NCOLP;
    Ag[(size_t)r * LDA + c] = (r == c) ? 1.0f : 0.0f;
  }
}

// --------------------------------------------------------------------------
// Kernel 6: blocked (NB=16) unpivoted LU, one workgroup per molecule.
//   panel -> L11^{-1} A12 triangular solve -> TDM stages U12 into LDS
//   -> WMMA trailing update.  Columns run to 544 so the b column (528)
//   is forward-eliminated in place.
// --------------------------------------------------------------------------
__global__ void k_lu(float* __restrict__ wsAaug) {
  int mol = blockIdx.x;
  float* Ag = wsAaug + (size_t)mol * NPAD * LDA;
  int tid = threadIdx.x, nthr = blockDim.x;   // 256, 8 waves
  int wave = tid >> 5, l = tid & 31, lm = l & 15, half = l >> 4;
  __shared__ float sUf[16 * 528];              // packed U12 block row (~34 KB)

  for (int kb = 0; kb < NPAD / 16; ++kb) {
    int kc = kb * 16;
    // ---- panel factorization on columns [kc, kc+16), rows [kc, NPAD) ----
    for (int j = 0; j < 16; ++j) {
      __syncthreads();
      int k = kc + j;
      float inv = 1.0f / Ag[(size_t)k * LDA + k];
      float pr[16];
      for (int jj = j + 1; jj < 16; ++jj) pr[jj] = Ag[(size_t)k * LDA + kc + jj];
      for (int i = k + 1 + tid; i < NPAD; i += nthr) {
        float lik = Ag[(size_t)i * LDA + k] * inv;
        Ag[(size_t)i * LDA + k] = lik;
        for (int jj = j + 1; jj < 16; ++jj)
          Ag[(size_t)i * LDA + kc + jj] -= lik * pr[jj];
      }
    }
    __syncthreads();
    int base = kc + 16;
    // ---- U12 = L11^{-1} A12 over columns [base, NCOLP) ----
    for (int c = base + tid; c < NCOLP; c += nthr) {
      float y[16];
      #pragma unroll
      for (int jj = 0; jj < 16; ++jj) y[jj] = Ag[(size_t)(kc + jj) * LDA + c];
      for (int jj = 1; jj < 16; ++jj) {
        float s = y[jj];
        for (int tt = 0; tt < jj; ++tt)
          s -= Ag[(size_t)(kc + jj) * LDA + kc + tt] * y[tt];
        y[jj] = s;
      }
      #pragma unroll
      for (int jj = 0; jj < 16; ++jj) Ag[(size_t)(kc + jj) * LDA + c] = y[jj];
    }
    __syncthreads();
    int nc = NCOLP - base;                     // <= 528
    // ---- TDM: DMA the 16 x nc U12 block row (stride LDA) packed into LDS.
    // D# descriptor per CDNA5 ISA (2D tensor, groups
    // 2/3 unused/zero). Issued once (wave 0); tracked with TENSORcnt.
    if (wave == 0) {
      uint64_t ga = (uint64_t)(uintptr_t)(Ag + (size_t)kc * LDA + base);
      v4u g0;
      g0[0] = 1u;                                    // count=1, user desc
      g0[1] = (uint32_t)(uintptr_t)sUf;              // lds_addr (LDS offset)
      g0[2] = (uint32_t)ga;                          // global_addr[31:0]
      g0[3] = (uint32_t)((ga >> 32) & 0x1FFFFFFu)    // global_addr[56:32]
              | (2u << 30);                          // type=2 ("image")
      v8i g1;
      g1[0] = (int)(2u << 16);                       // data_size=4B; mask=0
      g1[1] = (int)(((uint32_t)nc & 0xFFFFu) << 16); // tensor_dim0 lo16
      g1[2] = (int)(16u << 16);                      // dim0 hi=0 | tensor_dim1=16
      g1[3] = (int)(((uint32_t)nc & 0xFFFFu) << 16); // dim1 hi=0 | tile_dim0=nc
      g1[4] = 16;                                    // tile_dim1=16, tile_dim2=0
      g1[5] = LDA;                                   // tensor_dim0_stride lo32
      g1[6] = 0;                                     // stride hi | dim1_stride lo
      g1[7] = 0;
      v4i gz4 = {0, 0, 0, 0};
      v8i gz8 = {0, 0, 0, 0, 0, 0, 0, 0};
      __builtin_amdgcn_tensor_load_to_lds(g0, g1, gz4, gz4, gz8, 0);
      __builtin_amdgcn_s_wait_tensorcnt(0);
    }
    // prefetch next diagonal block while waiting
    if (kb + 1 < NPAD / 16)
      __builtin_prefetch(Ag + (size_t)base * LDA + base, 0, 1);
    __syncthreads();
    // ---- trailing update: A22 -= L21 @ U12 via WMMA f32 16x16x4 ----
    int ntr = (NPAD - base) / 16;
    int ntc = nc / 16;
    for (int t = wave; t < ntr * ntc; t += (nthr >> 5)) {
      int i0 = base + (t / ntc) * 16;
      int c0r = (t % ntc) * 16;                // relative to `base`
      v8f acc = {0.f,0.f,0.f,0.f,0.f,0.f,0.f,0.f};
      #pragma unroll
      for (int s = 0; s < 4; ++s) {
        int k0 = s * 4 + half * 2;
        v2f a, b;
        a.x = Ag[(size_t)(i0 + lm) * LDA + kc + k0];
        a.y = Ag[(size_t)(i0 + lm) * LDA + kc + k0 + 1];
        b.x = sUf[k0 * nc + c0r + lm];
        b.y = sUf[(k0 + 1) * nc + c0r + lm];
        acc = __builtin_amdgcn_wmma_f32_16x16x4_f32(false, a, false, b,
                                                    (short)0, acc, false, false);
      }
      #pragma unroll
      for (int v = 0; v < 8; ++v) {
        int r = i0 + half * 8 + v, c = base + c0r + lm;
        Ag[(size_t)r * LDA + c] -= acc[v];
      }
    }
    __syncthreads();
  }
}

// --------------------------------------------------------------------------
// Kernel 7: back-substitution U x = y (y = eliminated b column, col 528).
// --------------------------------------------------------------------------
__global__ void k_backsub(const float* __restrict__ wsAaug,
                          float* __restrict__ wsX) {
  int mol = blockIdx.x;
  const float* Ag = wsAaug + (size_t)mol * NPAD * LDA;
  int tid = threadIdx.x, nthr = blockDim.x;   // 512
  __shared__ float yv[NPAD];
  __shared__ float xi_s;
  for (int r = tid; r < NPAD; r += nthr) yv[r] = Ag[(size_t)r * LDA + 528];
  __syncthreads();
  for (int i = NPAD - 1; i >= 0; --i) {
    if (tid == 0) xi_s = yv[i] / Ag[(size_t)i * LDA + i];
    __syncthreads();
    float xi = xi_s;
    for (int r = tid; r < i; r += nthr) yv[r] -= Ag[(size_t)r * LDA + i] * xi;
    if (tid == 0) yv[i] = xi;
    __syncthreads();
  }
  for (int r = tid; r < EE; r += nthr) wsX[(size_t)mol * EE + r] = yv[r];
}

// --------------------------------------------------------------------------
// Kernel 8: charges = T @ x  -> out [B*N].
// --------------------------------------------------------------------------
__global__ void k_final(const float* __restrict__ T,
                        const float* __restrict__ wsX,
                        float* __restrict__ out) {
  int mol = blockIdx.x;
  int n = threadIdx.x;                  // 64
  __shared__ float xs[EE];
  for (int e = n; e < EE; e += NN) xs[e] = wsX[(size_t)mol * EE + e];
  __syncthreads();
  const float* Tm = T + (size_t)mol * NN * EE;
  float acc = 0.0f;
  #pragma unroll 4
  for (int e = 0; e < EE; ++e) acc += Tm[(size_t)n * EE + e] * xs[e];
  out[(size_t)mol * NN + n] = acc;
}

// --------------------------------------------------------------------------
extern "C" void kernel_launch(void* const* d_in, const int* in_sizes, int n_in,
                              void* d_out, int out_size, void* d_ws, size_t ws_size,
                              hipStream_t stream) {
  const float* positions  = (const float*)d_in[0];  // [B,64,3]
  const float* T          = (const float*)d_in[1];  // [B,64,512]
  const float* eneg       = (const float*)d_in[2];  // [B,64]
  const float* node_attrs = (const float*)d_in[3];  // [B,64,10]
  const float* qtot       = (const float*)d_in[4];  // [B]
  // d_in[5] = p (unused)
  const float* hardness   = (const float*)d_in[6];  // [10]
  const float* cov_radii  = (const float*)d_in[7];  // [119]
  const int*   an         = (const int*)d_in[8];    // [B,64]
  const int*   eidx       = (const int*)d_in[9];    // [B,2,512]
  float* out = (float*)d_out;                       // [B*64]

  // workspace carve-up (floats); total ~318 MB (Aaug dominates)
  float* w       = (float*)d_ws;
  float* wsA     = w;                                      // B*64*64
  float* wsM     = wsA    + (size_t)BATCH * NN * NN;       // B*64*512
  float* wsCut   = wsM    + (size_t)BATCH * NN * EE;       // B*512
  float* wsBvec  = wsCut  + (size_t)BATCH * EE;            // B*512
  float* wsX     = wsBvec + (size_t)BATCH * EE;            // B*512
  float* wsAaug  = wsX    + (size_t)BATCH * EE;            // B*528*544
  (void)in_sizes; (void)n_in; (void)out_size; (void)ws_size;

  k_build_A   <<<BATCH, 256, 0, stream>>>(positions, node_attrs, an, hardness,
                                          cov_radii, wsA);
  k_edge_b    <<<BATCH, 512, 0, stream>>>(positions, T, eneg, eidx, wsCut, wsBvec);
  k_gemm_AT   <<<BATCH, 256, 0, stream>>>(T, wsA, wsM);
  k_gemm_TtM  <<<dim3(BATCH, 4), 256, 0, stream>>>(T, wsM, wsCut, wsAaug);
  k_finish_aaug<<<BATCH, 256, 0, stream>>>(wsBvec, qtot, wsAaug);
  k_lu        <<<BATCH, 256, 0, stream>>>(wsAaug);
  k_backsub   <<<BATCH, 512, 0, stream>>>(wsAaug, wsX);
  k_final     <<<BATCH, NN, 0, stream>>>(T, wsX, out);
}